// GraphEncoder_57956288692676
// MI455X (gfx1250) — compile-verified
//
#include <hip/hip_runtime.h>

// ---------------------------------------------------------------------------
// GAT (3 layers, heads=1) for MI455X / gfx1250.
// Memory-bound: ~3 GB of edge gather/scatter traffic dominates (~130us floor
// @ 23.3 TB/s). Dense h = x@W uses full-precision V_WMMA_F32_16X16X4_F32
// (fp32 matrix path is free relative to the scatter traffic).
// ---------------------------------------------------------------------------

#define N_NODES   100000
#define N_EDGES   1600000
#define N_GRAPHS  64
#define DIM       64
#define N_LAYERS  3
#define NEG_SLOPE 0.2f

typedef __attribute__((ext_vector_type(2))) float v2f;
typedef __attribute__((ext_vector_type(8))) float v8f;

__device__ __forceinline__ float lrelu(float v) {
    return v > 0.0f ? v : NEG_SLOPE * v;
}

// float atomic max via sign-aware int/uint trick (init with -inf)
__device__ __forceinline__ void atomicMaxF(float* addr, float v) {
    if (v >= 0.0f) atomicMax((int*)addr, __float_as_int(v));
    else           atomicMin((unsigned int*)addr, __float_as_uint(v));
}

__device__ __forceinline__ void edge_nodes(const int* __restrict__ ei, int e,
                                           int& s, int& d) {
    if (e < N_EDGES) { s = ei[e]; d = ei[N_EDGES + e]; }
    else             { s = e - N_EDGES; d = s; }   // self-loop
}

// ---------------------------------------------------------------------------
// h = X @ W  (WMMA fp32), plus alpha_s = h.a_src, alpha_d = h.a_dst
// block = 128 threads = 4 waves; each wave makes a 16x64 tile of h.
// A frag (16x4 f32, ISA 7.12.2): lanes 0-15 hold K{0,1}, lanes 16-31 K{2,3}.
// B frag (4x16 f32):             VGPR0: K0 / K2 per half-wave, VGPR1: K1 / K3.
// ---------------------------------------------------------------------------
__global__ __launch_bounds__(128) void gat_gemm_alpha(
    const float* __restrict__ X, const float* __restrict__ W,
    const float* __restrict__ a_s, const float* __restrict__ a_d,
    float* __restrict__ H, float* __restrict__ AS, float* __restrict__ AD,
    int relu_in)
{
    __shared__ float lW[DIM * 65];   // padded stride 65 -> conflict-free
    __shared__ float lH[64 * 65];
    __shared__ float lAS[DIM], lAD[DIM];

    const int tid = threadIdx.x;
    for (int i = tid; i < DIM * DIM; i += 128)
        lW[(i >> 6) * 65 + (i & 63)] = W[i];
    if (tid < DIM) { lAS[tid] = a_s[tid]; lAD[tid] = a_d[tid]; }
    __syncthreads();

    const int wave    = tid >> 5;
    const int lane    = tid & 31;
    const int halfsel = lane >> 4;          // 0: lanes 0-15, 1: lanes 16-31
    const int l16     = lane & 15;
    const int rowBase = blockIdx.x * 64 + wave * 16;
    int row = rowBase + l16;
    if (row >= N_NODES) row = N_NODES - 1;  // keep EXEC full for WMMA
    const int koff = halfsel * 2;

    v8f acc0 = {}, acc1 = {}, acc2 = {}, acc3 = {};
    const float* xrow = X + (size_t)row * DIM;

#pragma unroll
    for (int kk = 0; kk < 16; ++kk) {
        const int k = kk * 4 + koff;
        float2 av = *(const float2*)(xrow + k);
        if (relu_in) { av.x = fmaxf(av.x, 0.0f); av.y = fmaxf(av.y, 0.0f); }
        v2f a; a.x = av.x; a.y = av.y;

        const float* w0 = lW + k * 65;
        const float* w1 = lW + (k + 1) * 65;
        v2f b0, b1, b2, b3;
        b0.x = w0[ 0 + l16]; b0.y = w1[ 0 + l16];
        b1.x = w0[16 + l16]; b1.y = w1[16 + l16];
        b2.x = w0[32 + l16]; b2.y = w1[32 + l16];
        b3.x = w0[48 + l16]; b3.y = w1[48 + l16];

        acc0 = __builtin_amdgcn_wmma_f32_16x16x4_f32(false, a, false, b0, (short)0, acc0, false, false);
        acc1 = __builtin_amdgcn_wmma_f32_16x16x4_f32(false, a, false, b1, (short)0, acc1, false, false);
        acc2 = __builtin_amdgcn_wmma_f32_16x16x4_f32(false, a, false, b2, (short)0, acc2, false, false);
        acc3 = __builtin_amdgcn_wmma_f32_16x16x4_f32(false, a, false, b3, (short)0, acc3, false, false);
    }

    // C/D layout: VGPR r -> row r (lanes 0-15) / row r+8 (lanes 16-31)
#pragma unroll
    for (int r = 0; r < 8; ++r) {
        const int m    = r + halfsel * 8;
        const int grow = rowBase + m;
        const int lrow = wave * 16 + m;
        const float c0 = acc0[r], c1 = acc1[r], c2 = acc2[r], c3 = acc3[r];
        lH[lrow * 65 +  0 + l16] = c0;
        lH[lrow * 65 + 16 + l16] = c1;
        lH[lrow * 65 + 32 + l16] = c2;
        lH[lrow * 65 + 48 + l16] = c3;
        if (grow < N_NODES) {
            float* hp = H + (size_t)grow * DIM;
            hp[ 0 + l16] = c0;
            hp[16 + l16] = c1;
            hp[32 + l16] = c2;
            hp[48 + l16] = c3;
        }
    }
    __syncthreads();

    // per-node attention coefficients from the LDS tile
    if (tid < 64) {
        const int grow = blockIdx.x * 64 + tid;
        if (grow < N_NODES) {
            float s = 0.0f, d = 0.0f;
            const float* hr = lH + tid * 65;
#pragma unroll
            for (int c = 0; c < DIM; ++c) {
                const float v = hr[c];
                s += v * lAS[c];
                d += v * lAD[c];
            }
            AS[grow] = s;
            AD[grow] = d;
        }
    }
}

// out := bias (aggregation accumulates on top), m := -inf, sum := 0
__global__ void gat_init(float* __restrict__ Hout, const float* __restrict__ bias,
                         float* __restrict__ M, float* __restrict__ S)
{
    const int i = blockIdx.x * blockDim.x + threadIdx.x;
    if (i < N_NODES * DIM) {
        Hout[i] = bias[i & (DIM - 1)];
        if ((i & (DIM - 1)) == 0) {
            const int n = i >> 6;
            M[n] = -__builtin_inff();
            S[n] = 0.0f;
        }
    }
}

__global__ void gat_edge_max(const int* __restrict__ ei, const float* __restrict__ AS,
                             const float* __restrict__ AD, float* __restrict__ M)
{
    const int e = blockIdx.x * blockDim.x + threadIdx.x;
    if (e >= N_EDGES + N_NODES) return;
    int s, d; edge_nodes(ei, e, s, d);
    atomicMaxF(&M[d], lrelu(AS[s] + AD[d]));
}

__global__ void gat_edge_sum(const int* __restrict__ ei, const float* __restrict__ AS,
                             const float* __restrict__ AD, const float* __restrict__ M,
                             float* __restrict__ S)
{
    const int e = blockIdx.x * blockDim.x + threadIdx.x;
    if (e >= N_EDGES + N_NODES) return;
    int s, d; edge_nodes(ei, e, s, d);
    const float ev = lrelu(AS[s] + AD[d]);
    atomicAdd(&S[d], __expf(ev - M[d]));
}

// 16 lanes per edge, float4 gather of h[src], scatter-add to out[dst]
__global__ __launch_bounds__(256) void gat_edge_aggregate(
    const int* __restrict__ ei, const float* __restrict__ AS,
    const float* __restrict__ AD, const float* __restrict__ M,
    const float* __restrict__ S, const float* __restrict__ H,
    float* __restrict__ Hout)
{
    const int t    = blockIdx.x * 256 + threadIdx.x;
    const int e    = t >> 4;
    const int lane = t & 15;
    if (e >= N_EDGES + N_NODES) return;
    int s, d; edge_nodes(ei, e, s, d);
    const float ev    = lrelu(AS[s] + AD[d]);
    const float alpha = __expf(ev - M[d]) / S[d];
    const float4 v    = ((const float4*)(H + (size_t)s * DIM))[lane];
    float* op = Hout + (size_t)d * DIM + lane * 4;
    atomicAdd(op + 0, v.x * alpha);
    atomicAdd(op + 1, v.y * alpha);
    atomicAdd(op + 2, v.z * alpha);
    atomicAdd(op + 3, v.w * alpha);
}

__global__ void pool_init(float* __restrict__ ps, float* __restrict__ pc)
{
    const int i = blockIdx.x * blockDim.x + threadIdx.x;
    if (i < N_GRAPHS * DIM) ps[i] = 0.0f;
    if (i < N_GRAPHS)       pc[i] = 0.0f;
}

// batch_ids are sorted -> per-block LDS accumulation, then one global flush
__global__ __launch_bounds__(256) void pool_accumulate(
    const float* __restrict__ H, const int* __restrict__ batch,
    float* __restrict__ ps, float* __restrict__ pc)
{
    __shared__ float lsum[N_GRAPHS * DIM];
    __shared__ float lcnt[N_GRAPHS];
    const int tid = threadIdx.x;
    for (int i = tid; i < N_GRAPHS * DIM; i += 256) lsum[i] = 0.0f;
    if (tid < N_GRAPHS) lcnt[tid] = 0.0f;
    __syncthreads();

    const int CHUNK = 1024;
    const int base  = blockIdx.x * CHUNK;
    for (int i = tid; i < CHUNK * DIM; i += 256) {
        const int node = base + (i >> 6);
        if (node < N_NODES) {
            const int c = i & 63;
            const int g = batch[node];
            atomicAdd(&lsum[g * DIM + c], H[(size_t)node * DIM + c]);
            if (c == 0) atomicAdd(&lcnt[g], 1.0f);
        }
    }
    __syncthreads();

    for (int i = tid; i < N_GRAPHS * DIM; i += 256) atomicAdd(&ps[i], lsum[i]);
    if (tid < N_GRAPHS) atomicAdd(&pc[tid], lcnt[tid]);
}

__global__ void pool_finalize(const float* __restrict__ ps, const float* __restrict__ pc,
                              float* __restrict__ out)
{
    const int i = blockIdx.x * blockDim.x + threadIdx.x;
    if (i < N_GRAPHS * DIM) {
        const int g = i >> 6;
        out[i] = ps[i] / fmaxf(pc[g], 1.0f);
    }
}

extern "C" void kernel_launch(void* const* d_in, const int* in_sizes, int n_in,
                              void* d_out, int out_size, void* d_ws, size_t ws_size,
                              hipStream_t stream)
{
    (void)in_sizes; (void)n_in; (void)out_size; (void)ws_size;

    const float* x   = (const float*)d_in[0];
    const int*   ei  = (const int*)d_in[1];
    const int*   bat = (const int*)d_in[2];
    const float* W   = (const float*)d_in[3];
    const float* a_s = (const float*)d_in[4];
    const float* a_d = (const float*)d_in[5];
    const float* b   = (const float*)d_in[6];
    float* out = (float*)d_out;

    char* ws = (char*)d_ws;
    size_t off = 0;
    auto carve = [&](size_t bytes) -> void* {
        void* p = ws + off;
        off = (off + bytes + 255) & ~(size_t)255;
        return p;
    };
    float* F    = (float*)carve((size_t)N_NODES * DIM * sizeof(float)); // h = X@W
    float* Hout = (float*)carve((size_t)N_NODES * DIM * sizeof(float)); // layer output
    float* AS   = (float*)carve((size_t)N_NODES * sizeof(float));
    float* AD   = (float*)carve((size_t)N_NODES * sizeof(float));
    float* M    = (float*)carve((size_t)N_NODES * sizeof(float));
    float* S    = (float*)carve((size_t)N_NODES * sizeof(float));
    float* PS   = (float*)carve((size_t)N_GRAPHS * DIM * sizeof(float));
    float* PC   = (float*)carve((size_t)N_GRAPHS * sizeof(float));

    const int ET = N_EDGES + N_NODES;
    const float* Xl = x;
    for (int l = 0; l < N_LAYERS; ++l) {
        gat_gemm_alpha<<<(N_NODES + 63) / 64, 128, 0, stream>>>(
            Xl, W + l * DIM * DIM, a_s + l * DIM, a_d + l * DIM,
            F, AS, AD, l > 0 ? 1 : 0);
        gat_init<<<(N_NODES * DIM + 255) / 256, 256, 0, stream>>>(Hout, b + l * DIM, M, S);
        gat_edge_max<<<(ET + 255) / 256, 256, 0, stream>>>(ei, AS, AD, M);
        gat_edge_sum<<<(ET + 255) / 256, 256, 0, stream>>>(ei, AS, AD, M, S);
        gat_edge_aggregate<<<((size_t)ET * 16 + 255) / 256, 256, 0, stream>>>(
            ei, AS, AD, M, S, F, Hout);
        Xl = Hout;
    }
    pool_init<<<(N_GRAPHS * DIM + 255) / 256, 256, 0, stream>>>(PS, PC);
    pool_accumulate<<<(N_NODES + 1023) / 1024, 256, 0, stream>>>(Hout, bat, PS, PC);
    pool_finalize<<<(N_GRAPHS * DIM + 255) / 256, 256, 0, stream>>>(PS, PC, out);
}